// BaseModel_58136677319062
// MI455X (gfx1250) — compile-verified
//
#include <hip/hip_runtime.h>
#include <hip/hip_bf16.h>

// GraphSAGE-mean x2 on gfx1250 (MI455X).
// FP32 WMMA (16x16x4) for exact-precision GEMMs; flat atomic scatter for the
// edge aggregation (memory-bound part, ~0.9 GB of HBM/L2 traffic per pass).

typedef __attribute__((ext_vector_type(2))) float v2f;
typedef __attribute__((ext_vector_type(8))) float v8f;

#define FEAT 128
#define ROWS_PER_WAVE 32    // 2 M-tiles per wave -> 2x B-fragment reuse
#define TILE_ROWS 256       // rows of output per block (8 waves x 32 rows)

// ---------------------------------------------------------------- zero (f4)
__global__ __launch_bounds__(256) void zero4_kernel(float4* __restrict__ p, long long n4) {
    long long i = (long long)blockIdx.x * 256 + threadIdx.x;
    if (i < n4) p[i] = make_float4(0.f, 0.f, 0.f, 0.f);
}

// ---------------------------------------------------------------- degrees
__global__ __launch_bounds__(256) void deg_kernel(const int* __restrict__ dst,
                                                  float* __restrict__ deg, int E) {
    int i = blockIdx.x * 256 + threadIdx.x;
    if (i < E) atomicAdd(&deg[dst[i]], 1.0f);
}

// -------------------------------------------------- edge scatter: agg[dst]+=h[src]
// One wave per edge: lane handles 4 consecutive floats (float4 gather read is a
// fully-coalesced 512B row (4x 128B lines) per wave; atomics land per-dword in L2).
__global__ __launch_bounds__(256) void scatter_kernel(const float* __restrict__ h,
                                                      const int* __restrict__ src,
                                                      const int* __restrict__ dst,
                                                      float* __restrict__ agg, int E) {
    int i = blockIdx.x * 256 + threadIdx.x;   // i in [0, E*32)
    int e = i >> 5;
    if (e >= E) return;
    int q = (i & 31) * 4;
    int s = src[e];
    int d = dst[e];
    float4 v = *(const float4*)(h + (size_t)s * FEAT + q);
    float* p = agg + (size_t)d * FEAT + q;
    atomicAdd(p + 0, v.x);
    atomicAdd(p + 1, v.y);
    atomicAdd(p + 2, v.z);
    atomicAdd(p + 3, v.w);
}

// -------------------------------------------------- fused SAGE layer GEMM
// out = relu( hself @ Wself + (agg/max(deg,1)) @ Wneigh + bias )
// Block: 256 threads = 8 waves; each wave owns a 32x128 output strip
// (2 M-tiles x 8 N-tiles of 16x16, v8f each = 128 acc VGPRs).
// Two phases ([h|W_self] then [mean|W_neigh]) share one 64KB LDS weight buffer.
__global__ __launch_bounds__(256)
void sage_gemm_kernel(const float* __restrict__ hself,
                      const float* __restrict__ agg,
                      const float* __restrict__ deg,
                      const float* __restrict__ Wself,
                      const float* __restrict__ Wneigh,
                      const float* __restrict__ bias,
                      float* __restrict__ out,
                      int M) {
    // Swizzled weight tile: element W[k][c] lives at sW[(k>>2)*512 + c*4 + (k&3)]
    // -> per-lane B fragment (k0+2h, k0+2h+1) is one 8-byte conflict-free load.
    __shared__ float sW[FEAT * FEAT];

    const int tid  = threadIdx.x;
    const int wave = tid >> 5;
    const int lane = tid & 31;
    const int half = lane >> 4;      // 0: lanes 0-15, 1: lanes 16-31
    const int l16  = lane & 15;

    const int rbase = blockIdx.x * TILE_ROWS + wave * ROWS_PER_WAVE;
    const int arow0 = rbase + l16;             // A row, M-tile 0
    const int arow1 = arow0 + 16;              // A row, M-tile 1
    const int ac0 = (arow0 < M) ? arow0 : (M - 1);  // clamp loads; stores predicated
    const int ac1 = (arow1 < M) ? arow1 : (M - 1);

    // inverse mean degree (identical for both half-lanes of a row)
    const float invd0 = 1.0f / fmaxf(deg[ac0], 1.0f);
    const float invd1 = 1.0f / fmaxf(deg[ac1], 1.0f);

    v8f acc[2][8] = {};

    #pragma unroll 1
    for (int phase = 0; phase < 2; ++phase) {
        const float* W = phase ? Wneigh : Wself;
        const float* A = phase ? agg : hself;
        const float  s0 = phase ? invd0 : 1.0f;
        const float  s1 = phase ? invd1 : 1.0f;

        __syncthreads();  // previous phase done reading sW
        // cooperative swizzled load of 128x128 fp32 weights (float4 global reads)
        for (int i = tid; i < (FEAT * FEAT) / 4; i += 256) {
            int k = i >> 5;            // row
            int c = (i & 31) * 4;      // col base
            float4 w = ((const float4*)W)[i];
            int base = (k >> 2) * 512 + (k & 3);
            sW[base + (c + 0) * 4] = w.x;
            sW[base + (c + 1) * 4] = w.y;
            sW[base + (c + 2) * 4] = w.z;
            sW[base + (c + 3) * 4] = w.w;
        }
        __syncthreads();

        const float* A0 = A + (size_t)ac0 * FEAT + 2 * half;
        const float* A1 = A + (size_t)ac1 * FEAT + 2 * half;

        // software pipeline, distance 2: the A fragments consumed at step ks
        // were issued at step ks-2, so a full loop body (4 DS loads + 16
        // WMMAs) of independent work covers the WGP$/L2 latency even after
        // the scheduler rotates the loop.
        v2f a0p0 = *(const v2f*)(A0);
        v2f a1p0 = *(const v2f*)(A1);
        v2f a0p1 = *(const v2f*)(A0 + 4);
        v2f a1p1 = *(const v2f*)(A1 + 4);

        #pragma unroll 2
        for (int ks = 0; ks < FEAT / 4; ++ks) {
            v2f a0 = a0p0, a1 = a1p0;
            a0p0 = a0p1; a1p0 = a1p1;
            int ksn = (ks + 2 < FEAT / 4) ? (ks + 2) : (FEAT / 4 - 1);
            a0p1 = *(const v2f*)(A0 + ksn * 4);
            a1p1 = *(const v2f*)(A1 + ksn * 4);
            a0.x *= s0; a0.y *= s0;
            a1.x *= s1; a1.y *= s1;

            // stage all 8 B fragments first: 4x ds_load_2addr_b64 issued as a
            // batch, staggered dscnt waits interleave with the WMMA stream.
            v2f b[8];
            const float* bptr = &sW[ks * 512 + 2 * half];
            #pragma unroll
            for (int n = 0; n < 8; ++n)
                b[n] = *(const v2f*)(bptr + (n * 16 + l16) * 4);

            #pragma unroll
            for (int n = 0; n < 8; ++n) {
                acc[0][n] = __builtin_amdgcn_wmma_f32_16x16x4_f32(
                    false, a0, false, b[n], (short)0, acc[0][n], false, false);
                acc[1][n] = __builtin_amdgcn_wmma_f32_16x16x4_f32(
                    false, a1, false, b[n], (short)0, acc[1][n], false, false);
            }
        }
    }

    // bias per lane-column (8 values)
    float bv[8];
    #pragma unroll
    for (int n = 0; n < 8; ++n) bv[n] = bias[n * 16 + l16];

    // Epilogue: C tile layout -> row = tilebase + r + 8*half, col = n*16 + l16
    #pragma unroll
    for (int t = 0; t < 2; ++t) {
        const int orow0 = rbase + t * 16 + 8 * half;
        #pragma unroll
        for (int r = 0; r < 8; ++r) {
            int row = orow0 + r;
            if (row < M) {
                #pragma unroll
                for (int n = 0; n < 8; ++n) {
                    int col = n * 16 + l16;
                    float v = acc[t][n][r] + bv[n];
                    out[(size_t)row * FEAT + col] = v > 0.f ? v : 0.f;
                }
            }
        }
    }
}

// ------------------------------------------------------------------ launch
extern "C" void kernel_launch(void* const* d_in, const int* in_sizes, int n_in,
                              void* d_out, int out_size, void* d_ws, size_t ws_size,
                              hipStream_t stream) {
    const float* x   = (const float*)d_in[0];
    const int*   src = (const int*)d_in[1];
    const int*   dst = (const int*)d_in[2];
    const float* Ws1 = (const float*)d_in[3];
    const float* Wn1 = (const float*)d_in[4];
    const float* b1  = (const float*)d_in[5];
    const float* Ws2 = (const float*)d_in[6];
    const float* Wn2 = (const float*)d_in[7];
    const float* b2  = (const float*)d_in[8];
    float* out = (float*)d_out;

    const int N = in_sizes[0] / FEAT;   // 100000 nodes
    const int E = in_sizes[1];          // 800000 edges

    float* agg = (float*)d_ws;                  // [N,128]
    float* deg = agg + (size_t)N * FEAT;        // [N]

    const int gemm_grid    = (N + TILE_ROWS - 1) / TILE_ROWS;
    const int scatter_grid = (E * 32 + 255) / 256;
    const int deg_grid     = (E + 255) / 256;

    // ---- layer 1 ----
    {   // zero agg + deg together (contiguous, N*129 floats, multiple of 4)
        long long n4 = ((long long)N * (FEAT + 1)) / 4;
        zero4_kernel<<<(int)((n4 + 255) / 256), 256, 0, stream>>>((float4*)agg, n4);
    }
    deg_kernel<<<deg_grid, 256, 0, stream>>>(dst, deg, E);
    scatter_kernel<<<scatter_grid, 256, 0, stream>>>(x, src, dst, agg, E);
    // hidden layer lives in d_out (each GEMM block reads only its own A rows
    // before writing them, so in-place layer-2 reuse is safe)
    sage_gemm_kernel<<<gemm_grid, 256, 0, stream>>>(x, agg, deg, Ws1, Wn1, b1, out, N);

    // ---- layer 2 ----
    {   long long n4 = ((long long)N * FEAT) / 4;
        zero4_kernel<<<(int)((n4 + 255) / 256), 256, 0, stream>>>((float4*)agg, n4);
    }
    scatter_kernel<<<scatter_grid, 256, 0, stream>>>(out, src, dst, agg, E);
    sage_gemm_kernel<<<gemm_grid, 256, 0, stream>>>(out, agg, deg, Ws2, Wn2, b2, out, N);
}